// VGAE_34660386078869
// MI455X (gfx1250) — compile-verified
//
#include <hip/hip_runtime.h>
#include <cstdint>

// ---------------------------------------------------------------------------
// VGAE forward (loss, acc) for MI455X / gfx1250, wave32.
// fp32 WMMA 16x16x4 + async global->LDS double-buffered staging.
// B=128, N=512, F=128, L=64, H=128, C=10
// ---------------------------------------------------------------------------

typedef float v2f __attribute__((ext_vector_type(2)));
typedef float v8f __attribute__((ext_vector_type(8)));
typedef int   v4i __attribute__((ext_vector_type(4)));
typedef __attribute__((address_space(1))) v4i* gas_v4i;   // non-const: builtin param is mutable
typedef __attribute__((address_space(3))) v4i* las_v4i;

#define Bv 128
#define Nv 512
#define Fv 128
#define Lv 64
#define Hv 128
#define Cv 10

#if defined(__has_builtin)
#if __has_builtin(__builtin_amdgcn_global_load_async_to_lds_b128) && \
    __has_builtin(__builtin_amdgcn_s_wait_asynccnt)
#define HAS_ASYNC_LDS 1
#endif
#endif
#ifndef HAS_ASYNC_LDS
#define HAS_ASYNC_LDS 0
#endif

__device__ __forceinline__ float eluf(float x) { return x > 0.f ? x : expm1f(x); }

// copy 16 aligned bytes global -> LDS (async DMA if available)
__device__ __forceinline__ void cp16(const float* gsrc, float* ldst) {
#if HAS_ASYNC_LDS
  __builtin_amdgcn_global_load_async_to_lds_b128(
      (gas_v4i)(uintptr_t)gsrc,
      (las_v4i)(uint32_t)(uintptr_t)ldst, 0, 0);
#else
  const float4 v = *(const float4*)gsrc;
  ldst[0] = v.x; ldst[1] = v.y; ldst[2] = v.z; ldst[3] = v.w;
#endif
}

__device__ __forceinline__ void wait_cp(int more_pending) {
#if HAS_ASYNC_LDS
  if (more_pending) __builtin_amdgcn_s_wait_asynccnt(3);  // drain current tile (3/wave, in-order)
  else              __builtin_amdgcn_s_wait_asynccnt(0);
#else
  (void)more_pending;
#endif
}

// ---------------------------------------------------------------------------
// K0: invdeg[b,j] = 1 / sum_i g[b,i,j]  (column sums), zero accumulators
// ---------------------------------------------------------------------------
__global__ void k_deg(const float* __restrict__ g, float* __restrict__ invdeg,
                      float* __restrict__ acc) {
  int idx = blockIdx.x * blockDim.x + threadIdx.x;  // b*N + j
  if (blockIdx.x == 0 && threadIdx.x < 8) acc[threadIdx.x] = 0.f;
  int b = idx >> 9, j = idx & (Nv - 1);
  const float* gb = g + (size_t)b * Nv * Nv + j;
  float s = 0.f;
  for (int i = 0; i < Nv; ++i) s += gb[(size_t)i * Nv];
  invdeg[idx] = 1.f / s;
}

// ---------------------------------------------------------------------------
// K1/K3: y[i] = x[i] * scale[i >> shift]
// ---------------------------------------------------------------------------
__global__ void k_scale(const float* __restrict__ x, const float* __restrict__ scale,
                        float* __restrict__ y, int shift, int n) {
  int i = blockIdx.x * 256 + threadIdx.x;
  if (i < n) y[i] = x[i] * scale[i >> shift];
}

// ---------------------------------------------------------------------------
// WMMA fp32 GEMM: C[z] = A[z](MxK) @ B[z](KxN), optional bias+ELU epilogue.
// 256 thr = 8 wave32; block tile 128x64; per-wave 32x32 (2x2 16x16 frags).
// Double-buffered LDS, tiles staged with GLOBAL_LOAD_ASYNC_TO_LDS_B128.
// K mult of 16, M mult of 128, N mult of 64 (true for all call sites).
// ---------------------------------------------------------------------------
__global__ __launch_bounds__(256) void k_wmma_gemm(
    const float* __restrict__ A, const float* __restrict__ B, float* __restrict__ C,
    int lda, int ldb, int ldc, int K,
    long sA, long sB, long sC,
    const float* __restrict__ bias, int doElu) {
  // row strides are multiples of 4 floats so every 16B chunk is 16B-aligned
  __shared__ __align__(16) float As[2][128][20];
  __shared__ __align__(16) float Bs[2][16][68];

  A += (long)blockIdx.z * sA;
  B += (long)blockIdx.z * sB;
  C += (long)blockIdx.z * sC;

  const int t = threadIdx.x;
  const int wave = t >> 5, lane = t & 31;
  const int half = lane >> 4, r = lane & 15;
  const int wm = wave >> 1, wn = wave & 1;  // 4x2 wave grid
  const int m_blk = blockIdx.x * 128;
  const int n_blk = blockIdx.y * 64;

  // per-thread staging coordinates (3 x 16B chunks per thread per tile)
  const int a_row0 = t >> 2,          a_cg0 = (t & 3) * 4;
  const int a_row1 = (t + 256) >> 2,  a_cg1 = ((t + 256) & 3) * 4;
  const int b_row  = t >> 4,          b_cg  = (t & 15) * 4;

  auto stage = [&](int k0, int buf) {
    cp16(A + (size_t)(m_blk + a_row0) * lda + k0 + a_cg0, &As[buf][a_row0][a_cg0]);
    cp16(A + (size_t)(m_blk + a_row1) * lda + k0 + a_cg1, &As[buf][a_row1][a_cg1]);
    cp16(B + (size_t)(k0 + b_row) * ldb + n_blk + b_cg,   &Bs[buf][b_row][b_cg]);
  };

  v8f acc[2][2];
#pragma unroll
  for (int i = 0; i < 2; ++i)
#pragma unroll
    for (int j = 0; j < 2; ++j)
#pragma unroll
      for (int e = 0; e < 8; ++e) acc[i][j][e] = 0.f;

  const int nT = K >> 4;
  stage(0, 0);  // prologue

  for (int i = 0; i < nT; ++i) {
    const int cur = i & 1;
    const int have_next = (i + 1 < nT);
    if (have_next) stage((i + 1) << 4, cur ^ 1);
    wait_cp(have_next);   // my wave's tile-i copies complete
    __syncthreads();      // everyone's copies visible

#pragma unroll
    for (int kk = 0; kk < 16; kk += 4) {
      v2f af[2], bf[2];
#pragma unroll
      for (int mt = 0; mt < 2; ++mt) {
        int mo = wm * 32 + mt * 16 + r;  // A row; lane K = 2*half + {0,1}
        af[mt].x = As[cur][mo][kk + 2 * half + 0];
        af[mt].y = As[cur][mo][kk + 2 * half + 1];
      }
#pragma unroll
      for (int nt = 0; nt < 2; ++nt) {
        int no = wn * 32 + nt * 16 + r;  // B col; lane K = 2*half + {0,1}
        bf[nt].x = Bs[cur][kk + 2 * half + 0][no];
        bf[nt].y = Bs[cur][kk + 2 * half + 1][no];
      }
#pragma unroll
      for (int mt = 0; mt < 2; ++mt)
#pragma unroll
        for (int nt = 0; nt < 2; ++nt)
          acc[mt][nt] = __builtin_amdgcn_wmma_f32_16x16x4_f32(
              false, af[mt], false, bf[nt], (short)0, acc[mt][nt], false, false);
    }
    __syncthreads();  // compute done before anyone restages this buffer
  }

  // epilogue + store: D row = v + 8*half, col = lane&15
#pragma unroll
  for (int mt = 0; mt < 2; ++mt) {
#pragma unroll
    for (int nt = 0; nt < 2; ++nt) {
      int col = n_blk + wn * 32 + nt * 16 + r;
      float bval = bias ? bias[col] : 0.f;
#pragma unroll
      for (int v = 0; v < 8; ++v) {
        int rowg = m_blk + wm * 32 + mt * 16 + v + 8 * half;
        float x = acc[mt][nt][v];
        if (doElu) x = eluf(x + bval);
        C[(size_t)rowg * ldc + col] = x;
      }
    }
  }
}

// ---------------------------------------------------------------------------
// K4: mu/lv = elu(Q.Wmu/Wlv + b), z, v = sigmoid(beta z), vd = v*invdeg,
//     KL partial -> atomicAdd(acc[0])
// ---------------------------------------------------------------------------
__global__ void k_latent(const float* __restrict__ Q,
                         const float* __restrict__ Wmu, const float* __restrict__ bmu,
                         const float* __restrict__ Wlv, const float* __restrict__ blv,
                         const float* __restrict__ eps, const float* __restrict__ beta,
                         const float* __restrict__ invdeg,
                         float* __restrict__ v, float* __restrict__ vd,
                         float* __restrict__ acc) {
  __shared__ float red[256];
  int row = blockIdx.x * 256 + threadIdx.x;
  const float* q = Q + (size_t)row * Lv;
  float sm = 0.f, sl = 0.f;
#pragma unroll 8
  for (int k = 0; k < Lv; ++k) { float qv = q[k]; sm += qv * Wmu[k]; sl += qv * Wlv[k]; }
  float mu = eluf(sm + bmu[0]);
  float lv = eluf(sl + blv[0]);
  float z = mu + eps[row] * expf(0.5f * lv);
  float vv = 1.f / (1.f + expf(-beta[0] * z));
  v[row] = vv;
  vd[row] = vv * invdeg[row];
  red[threadIdx.x] = 1.f + 2.f * lv - mu * mu - expf(2.f * lv);
  __syncthreads();
  for (int off = 128; off > 0; off >>= 1) {
    if (threadIdx.x < off) red[threadIdx.x] += red[threadIdx.x + off];
    __syncthreads();
  }
  if (threadIdx.x == 0) atomicAdd(&acc[0], red[0]);
}

// ---------------------------------------------------------------------------
// Ksv: s[b,i] = sum_m g[b,i,m] * vd[b,m]  (one wave32 per row, shfl reduce)
// ---------------------------------------------------------------------------
__global__ __launch_bounds__(256) void k_matvec_s(const float* __restrict__ g,
                                                  const float* __restrict__ vd,
                                                  float* __restrict__ s) {
  int wave = threadIdx.x >> 5, lane = threadIdx.x & 31;
  int bi = blockIdx.x * 8 + wave;  // b*N + i
  int b = bi >> 9, i = bi & (Nv - 1);
  const float* grow = g + (size_t)b * Nv * Nv + (size_t)i * Nv;
  const float* vb = vd + b * Nv;
  float a = 0.f;
#pragma unroll 4
  for (int m = lane; m < Nv; m += 32) a += grow[m] * vb[m];
#pragma unroll
  for (int off = 16; off > 0; off >>= 1) a += __shfl_down(a, off, 32);
  if (lane == 0) s[bi] = a;
}

// ---------------------------------------------------------------------------
// K5: hp[b,l] = sum_n h1[b,n,l] * v[b,n]
// ---------------------------------------------------------------------------
__global__ void k_pool(const float* __restrict__ h1, const float* __restrict__ v,
                       float* __restrict__ hp) {
  __shared__ float red[256];
  int b = blockIdx.x, t = threadIdx.x;
  int l = t & 63, q = t >> 6;
  float a = 0.f;
  for (int n = q; n < Nv; n += 4) a += h1[((size_t)b * Nv + n) * Lv + l] * v[b * Nv + n];
  red[t] = a;
  __syncthreads();
  if (q == 0) hp[b * Lv + l] = red[l] + red[l + 64] + red[l + 128] + red[l + 192];
}

// ---------------------------------------------------------------------------
// K6: mse partial = sum_l (h1 - elu(s*Wdec + bdec))^2 -> atomicAdd(acc[1])
// ---------------------------------------------------------------------------
__global__ void k_mse(const float* __restrict__ h1, const float* __restrict__ s,
                      const float* __restrict__ Wd, const float* __restrict__ bd,
                      float* __restrict__ acc) {
  __shared__ float red[256];
  int row = blockIdx.x * 256 + threadIdx.x;
  float sv = s[row];
  const float* hr = h1 + (size_t)row * Lv;
  float a = 0.f;
#pragma unroll 8
  for (int l = 0; l < Lv; ++l) {
    float d = eluf(sv * Wd[l] + bd[l]);
    float e = hr[l] - d;
    a += e * e;
  }
  red[threadIdx.x] = a;
  __syncthreads();
  for (int off = 128; off > 0; off >>= 1) {
    if (threadIdx.x < off) red[threadIdx.x] += red[threadIdx.x + off];
    __syncthreads();
  }
  if (threadIdx.x == 0) atomicAdd(&acc[1], red[0]);
}

// ---------------------------------------------------------------------------
// K7: MLP head + log_softmax + nll/acc, combine with kl/mse -> out[0..1]
// ---------------------------------------------------------------------------
__global__ __launch_bounds__(256) void k_head(
    const float* __restrict__ hp, const float* __restrict__ W1,
    const float* __restrict__ b1, const float* __restrict__ W2,
    const float* __restrict__ b2, const int* __restrict__ labels,
    const float* __restrict__ acc, float* __restrict__ out) {
  __shared__ float hps[Bv * Lv];
  __shared__ float rn[128];
  __shared__ float rc[128];
  int t = threadIdx.x;
  for (int i = t; i < Bv * Lv; i += 256) hps[i] = hp[i];
  __syncthreads();
  float nll_i = 0.f, cor = 0.f;
  if (t < Bv) {
    float lg[Cv];
#pragma unroll
    for (int j = 0; j < Cv; ++j) lg[j] = b2[j];
    for (int c = 0; c < Hv; ++c) {
      float a = b1[c];
#pragma unroll 8
      for (int k = 0; k < Lv; ++k) a += hps[t * Lv + k] * W1[k * Hv + c];
      a = eluf(a);
#pragma unroll
      for (int j = 0; j < Cv; ++j) lg[j] += a * W2[c * Cv + j];
    }
    float m = lg[0];
#pragma unroll
    for (int j = 1; j < Cv; ++j) m = fmaxf(m, lg[j]);
    float se = 0.f;
#pragma unroll
    for (int j = 0; j < Cv; ++j) se += expf(lg[j] - m);
    float lse = m + logf(se);
    int lab = labels[t];
    nll_i = -(lg[lab] - lse);
    int am = 0; float bvv = lg[0];
#pragma unroll
    for (int j = 1; j < Cv; ++j) if (lg[j] > bvv) { bvv = lg[j]; am = j; }
    cor = (am == lab) ? 1.f : 0.f;
  }
  if (t < 128) { rn[t] = nll_i; rc[t] = cor; }
  __syncthreads();
  for (int off = 64; off > 0; off >>= 1) {
    if (t < off) { rn[t] += rn[t + off]; rc[t] += rc[t + off]; }
    __syncthreads();
  }
  if (t == 0) {
    float nll = rn[0] / (float)Bv;
    const float M = (float)(Bv * Nv);
    float kl = -0.5f * acc[0] / (M * M);
    float mse = acc[1] / (M * (float)Lv);
    out[0] = nll + kl + mse;
    out[1] = rc[0] / (float)Bv;
  }
}

// ---------------------------------------------------------------------------
extern "C" void kernel_launch(void* const* d_in, const int* in_sizes, int n_in,
                              void* d_out, int out_size, void* d_ws, size_t ws_size,
                              hipStream_t stream) {
  const float* g     = (const float*)d_in[0];
  const float* h     = (const float*)d_in[1];
  const int*   labels= (const int*)d_in[2];
  const float* eps   = (const float*)d_in[3];
  const float* W_s   = (const float*)d_in[4];
  const float* b_s   = (const float*)d_in[5];
  const float* W_mu  = (const float*)d_in[6];
  const float* b_mu  = (const float*)d_in[7];
  const float* W_lv  = (const float*)d_in[8];
  const float* b_lv  = (const float*)d_in[9];
  const float* W_dec = (const float*)d_in[10];
  const float* b_dec = (const float*)d_in[11];
  const float* W1    = (const float*)d_in[12];
  const float* b1    = (const float*)d_in[13];
  const float* W2    = (const float*)d_in[14];
  const float* b2    = (const float*)d_in[15];
  const float* beta  = (const float*)d_in[16];
  float* out = (float*)d_out;
  float* ws  = (float*)d_ws;

  // workspace layout (floats); ~69 MB with lifetime-based reuse
  float* invdeg = ws;               // 65536
  float* v      = ws + 65536;       // 65536
  float* vd     = ws + 131072;      // 65536
  float* s      = ws + 196608;      // 65536
  float* hp     = ws + 262144;      // 8192
  float* accv   = ws + 270336;      // 8 (kl, mse sums)
  const size_t R1 = 524288;
  float* hs  = ws + R1;             // 8.4M  (dead after GEMM1)
  float* h1  = ws + R1;             // 4.2M  (reuses hs region)
  float* h1s = ws + R1 + 4194304;   // 4.2M
  const size_t R2 = R1 + 8388608;
  float* P = ws + R2;               // 8.4M  (dead after GEMM2)
  float* Q = ws + R2;               // 4.2M  (reuses P region)

  // 1) invdeg (column sums) + zero accumulators
  k_deg<<<256, 256, 0, stream>>>(g, invdeg, accv);
  // 2) hs = invdeg (*) h
  k_scale<<<(Bv * Nv * Fv) / 256, 256, 0, stream>>>(h, invdeg, hs, 7, Bv * Nv * Fv);
  // 3) P = g @ hs   [batched 512x512x128]
  k_wmma_gemm<<<dim3(Nv / 128, Fv / 64, Bv), 256, 0, stream>>>(
      g, hs, P, Nv, Fv, Fv, Nv,
      (long)Nv * Nv, (long)Nv * Fv, (long)Nv * Fv, nullptr, 0);
  // 4) h1 = elu(P @ W_s + b_s)   [65536x128x64]
  k_wmma_gemm<<<dim3((Bv * Nv) / 128, 1, 1), 256, 0, stream>>>(
      P, W_s, h1, Fv, Lv, Lv, Fv, 0, 0, 0, b_s, 1);
  // 5) h1s = invdeg (*) h1
  k_scale<<<(Bv * Nv * Lv) / 256, 256, 0, stream>>>(h1, invdeg, h1s, 6, Bv * Nv * Lv);
  // 6) Q = g @ h1s   [batched 512x512x64]
  k_wmma_gemm<<<dim3(Nv / 128, 1, Bv), 256, 0, stream>>>(
      g, h1s, Q, Nv, Lv, Lv, Nv,
      (long)Nv * Nv, (long)Nv * Lv, (long)Nv * Lv, nullptr, 0);
  // 7) mu/lv/z/v/vd + KL partial
  k_latent<<<(Bv * Nv) / 256, 256, 0, stream>>>(Q, W_mu, b_mu, W_lv, b_lv, eps, beta,
                                               invdeg, v, vd, accv);
  // 8) s = g @ vd (wave-per-row reduction)
  k_matvec_s<<<(Bv * Nv) / 8, 256, 0, stream>>>(g, vd, s);
  // 9) hp pooling
  k_pool<<<Bv, 256, 0, stream>>>(h1, v, hp);
  // 10) mse partial (fused decoder, d never materialized)
  k_mse<<<(Bv * Nv) / 256, 256, 0, stream>>>(h1, s, W_dec, b_dec, accv);
  // 11) head -> (loss, acc)
  k_head<<<1, 256, 0, stream>>>(hp, W1, b1, W2, b2, labels, accv, out);
}